// DenseSymmLayer_77137612636249
// MI455X (gfx1250) — compile-verified
//
#include <hip/hip_runtime.h>
#include <hip/hip_bf16.h>

// DenseSymm (cyclic-group equivariant dense layer) on gfx1250 via
// V_WMMA_F32_16X16X4_F32 (full f32 precision on the matrix pipe).
//
// out[b,m,g] = sum_{i,j} x[b,i,(j+shift_g)&1023] * K[m,i,j] + bias[m]
// shift_g = symm[g][0]  (symm is the cyclic translation group).
//
// Per batch b: C(64x1024) = A(64x4096) @ B(4096x1024), with B virtual:
// each row is a cyclic shift of x[b] staged *doubled* in LDS so the
// gather needs no wrap mask -> DS addresses are affine in the k loop
// (immediate offsets, ~zero per-step VALU).
//
// Workgroup = 1 batch x 128 g-columns, 8 waves as (wm 0..1)x(wn 0..3);
// each wave computes 32(M) x 32(g) of C as 2x2 WMMA fragments so both
// A and B fragments are reused by 2 WMMAs each.

#define BATCH    64
#define IN_FEAT  4
#define N_SITES  1024
#define N_SYMM   1024
#define FEATURES 64
#define KTOT     (IN_FEAT * N_SITES)   // 4096
#define GT       128                   // g-columns per workgroup
#define XS_ROW   (2 * N_SITES)         // doubled row, no wrap needed

typedef float v2f __attribute__((ext_vector_type(2)));
typedef float v8f __attribute__((ext_vector_type(8)));

__global__ __launch_bounds__(256)
void dense_symm_wmma_f32(const float* __restrict__ x,
                         const int*   __restrict__ symm,
                         const float* __restrict__ kern,
                         const float* __restrict__ bias,
                         float*       __restrict__ out)
{
    __shared__ float xs[IN_FEAT * XS_ROW];    // 32 KB: x[b,:,:] duplicated 2x
    __shared__ int   shifts[GT];              // symm[g][0] per g-column

    const int tid = threadIdx.x;
    const int b   = blockIdx.y;
    const int g0  = blockIdx.x * GT;

    // ---- stage x[b,:,:] doubled into LDS (coalesced float4) ----
    {
        const float4* src = (const float4*)(x + (size_t)b * (IN_FEAT * N_SITES));
        float4* dst = (float4*)xs;            // 2048 float4 slots
        #pragma unroll
        for (int t = 0; t < (IN_FEAT * XS_ROW / 4) / 256; ++t) {
            const int idx = tid + t * 256;    // 0..2047
            const int i   = idx >> 9;         // 512 float4 per doubled row
            const int c   = idx & 511;
            dst[idx] = src[(i << 8) + (c & 255)];   // wrap at staging time
        }
    }
    // ---- per-g shift from first column of symm (cyclic group) ----
    if (tid < GT)
        shifts[tid] = symm[(size_t)(g0 + tid) * N_SITES];

    __syncthreads();

    const int lane = tid & 31;
    const int w    = tid >> 5;      // 0..7
    const int wm   = w & 1;         // 32-row M half
    const int wn   = w >> 1;        // 0..3: 32-col g quarter
    const int l15  = lane & 15;
    const int h    = lane >> 4;     // lane half -> K offset 2h (ISA 7.12.2)
    const int m0   = wm * 32;
    const int gq   = wn * 32;

    // per-lane cyclic shifts for the 2 g-subtiles of this wave
    const int sh0 = shifts[gq + l15];
    const int sh1 = shifts[gq + 16 + l15];

    // accumulators pre-loaded with bias: C vgpr v <-> row (mtile + v + 8h)
    v8f c0, c1;
    #pragma unroll
    for (int v = 0; v < 8; ++v) {
        c0[v] = bias[m0 +      8 * h + v];
        c1[v] = bias[m0 + 16 + 8 * h + v];
    }
    v8f acc00 = c0, acc01 = c0;   // M tile 0 x {g sub 0, g sub 1}
    v8f acc10 = c1, acc11 = c1;   // M tile 1 x {g sub 0, g sub 1}

    // A fragment sources: lane l15 owns row, lane-half h owns K = 2h, 2h+1
    const float* Arow0 = kern + (size_t)(m0 + l15) * KTOT + 2 * h;
    const float* Arow1 = Arow0 + (size_t)16 * KTOT;

    #pragma unroll
    for (int i = 0; i < IN_FEAT; ++i) {
        // affine bases: no wrap mask thanks to the doubled LDS rows
        const float* xb0 = xs + i * XS_ROW + 2 * h + sh0;
        const float* xb1 = xs + i * XS_ROW + 2 * h + sh1;
        const float* Ai0 = Arow0 + i * N_SITES;
        const float* Ai1 = Arow1 + i * N_SITES;
        #pragma unroll 4
        for (int j = 0; j < N_SITES; j += 4) {
            // A fragments: 16x4 f32 tiles (global_load_b64, L2-hot)
            v2f a0 = *(const v2f*)(Ai0 + j);
            v2f a1 = *(const v2f*)(Ai1 + j);
            // B fragments: 4x16 cyclic-shift gathers (2x ds_load_b32 each,
            // addresses are base + immediate thanks to affine indexing)
            v2f b0, b1;
            b0[0] = xb0[j];  b0[1] = xb0[j + 1];
            b1[0] = xb1[j];  b1[1] = xb1[j + 1];

            acc00 = __builtin_amdgcn_wmma_f32_16x16x4_f32(
                        false, a0, false, b0, (short)0, acc00, false, false);
            acc01 = __builtin_amdgcn_wmma_f32_16x16x4_f32(
                        false, a0, false, b1, (short)0, acc01, false, false);
            acc10 = __builtin_amdgcn_wmma_f32_16x16x4_f32(
                        false, a1, false, b0, (short)0, acc10, false, false);
            acc11 = __builtin_amdgcn_wmma_f32_16x16x4_f32(
                        false, a1, false, b1, (short)0, acc11, false, false);
        }
    }

    // ---- epilogue: C vgpr v -> out[b, mtile + v + 8h, g] ----
    const int gc0 = g0 + gq + l15;
    const int gc1 = gc0 + 16;
    float* obase = out + ((size_t)b * FEATURES + m0 + 8 * h) * N_SYMM;
    #pragma unroll
    for (int v = 0; v < 8; ++v) {
        obase[(size_t)v * N_SYMM + gc0]        = acc00[v];
        obase[(size_t)v * N_SYMM + gc1]        = acc01[v];
        obase[(size_t)(v + 16) * N_SYMM + gc0] = acc10[v];
        obase[(size_t)(v + 16) * N_SYMM + gc1] = acc11[v];
    }
}

extern "C" void kernel_launch(void* const* d_in, const int* in_sizes, int n_in,
                              void* d_out, int out_size, void* d_ws, size_t ws_size,
                              hipStream_t stream) {
    (void)in_sizes; (void)n_in; (void)out_size; (void)d_ws; (void)ws_size;
    const float* x    = (const float*)d_in[0];
    const int*   symm = (const int*)d_in[1];
    const float* kern = (const float*)d_in[2];
    const float* bias = (const float*)d_in[3];
    float*       out  = (float*)d_out;

    dim3 grid(N_SYMM / GT, BATCH);   // 8 x 64 = 512 workgroups
    dim3 block(256);                 // 8 wave32s
    dense_symm_wmma_f32<<<grid, block, 0, stream>>>(x, symm, kern, bias, out);
}